// LearnedTimeAssymmetricDiffusion_31834297598779
// MI455X (gfx1250) — compile-verified
//
#include <hip/hip_runtime.h>

typedef __attribute__((ext_vector_type(2))) float v2f;
typedef __attribute__((ext_vector_type(8))) float v8f;

#define NB     4096      // batches
#define NN     87        // matrix dim
#define ROWS   96        // padded rows (6 * 16)
#define KP     88        // padded K (22 * 4)
#define STRIDE 92        // LDS row stride in floats (bank-conflict-free for b64 loads)
#define NTILES 6         // N tiles of 16
#define KSTEPS 22        // K steps of 4
#define THREADS 192      // 6 waves of 32

__launch_bounds__(THREADS)
__global__ void diffusion_kernel(const float* __restrict__ x,
                                 const float* __restrict__ evals,
                                 const float* __restrict__ evecs,
                                 const float* __restrict__ dtime,
                                 float* __restrict__ out)
{
    __shared__ __align__(16) float Esh[ROWS * STRIDE];  // evecs[b], zero-padded
    __shared__ float evs[KP];                           // evals[b], zero-padded
    __shared__ float tsh[ROWS];                         // clamped diffusion times

    const int b    = blockIdx.x;
    const int tid  = threadIdx.x;
    const int lane = tid & 31;
    const int wave = tid >> 5;           // 0..5, wave-uniform

    const float* __restrict__ evecs_b = evecs + (size_t)b * NN * NN;
    const float* __restrict__ evals_b = evals + (size_t)b * NN;
    const float* __restrict__ x_b     = x + (size_t)b * NN * NN;
    float* __restrict__ out_b         = out + (size_t)b * NN * (2 * NN);

    // ---- stage evals / diffusion_time -------------------------------------
    if (tid < KP) evs[tid] = (tid < NN) ? evals_b[tid] : 0.0f;
    if (tid >= KP && tid < KP + ROWS) {
        int n = tid - KP;
        float t = (n < NN) ? dtime[n] : 1.0f;
        tsh[n] = fmaxf(t, 1e-8f);        // torch clamp(min=1e-8)
    }
    // ---- stage evecs[b] into LDS with zero padding ------------------------
    for (int idx = tid; idx < ROWS * KP; idx += THREADS) {
        int n = idx / KP;
        int k = idx - n * KP;
        float v = 0.0f;
        if (n < NN && k < NN) v = evecs_b[n * NN + k];
        Esh[n * STRIDE + k] = v;
    }
    // ---- copy x into out[:, 0:87] while LDS fills -------------------------
    for (int idx = tid; idx < NN * NN; idx += THREADS) {
        int n = idx / NN;
        int m = idx - n * NN;
        out_b[n * (2 * NN) + m] = x_b[idx];
    }
    __syncthreads();

    // ---- per-wave A fragments in registers: A[n,k] = E[n,k]*exp(-t[n]*ev[k])
    // A 16x4 fp32 layout: lanes 0-15 hold M=lane (K=0,1); lanes 16-31 K=2,3.
    const int  rowA  = 16 * wave + (lane & 15);
    const int  khalf = (lane >> 4) * 2;          // 0 or 2
    const bool rowv  = rowA < NN;
    const float tn   = tsh[rowA];
    v2f afrag[KSTEPS];
#pragma unroll
    for (int kk = 0; kk < KSTEPS; ++kk) {
        const int k0 = 4 * kk + khalf;
        float e0 = Esh[rowA * STRIDE + k0];      // 0 for padded rows/cols
        float e1 = Esh[rowA * STRIDE + k0 + 1];
        float s0 = __expf(-tn * evs[k0]);        // v_exp_f32
        float s1 = __expf(-tn * evs[k0 + 1]);
        v2f a;
        a.x = rowv ? e0 * s0 : 0.0f;
        a.y = rowv ? e1 * s1 : 0.0f;
        afrag[kk] = a;
    }

    // ---- kernel = A @ E^T via V_WMMA_F32_16X16X4_F32 ----------------------
    // B 4x16 layout: lanes 0-15 hold N=lane (K=0,1); lanes 16-31 K=2,3.
    const int colB = lane & 15;
#pragma unroll
    for (int nt = 0; nt < NTILES; ++nt) {
        v8f acc = {};
        const float* bbase = &Esh[(nt * 16 + colB) * STRIDE + khalf];
#pragma unroll
        for (int kk = 0; kk < KSTEPS; ++kk) {
            v2f bfrag = *(const v2f*)(bbase + 4 * kk);   // ds_load_b64, conflict-free
            acc = __builtin_amdgcn_wmma_f32_16x16x4_f32(
                false, afrag[kk], false, bfrag,
                (short)0, acc, false, false);
        }
        // C/D layout: VGPR r -> M = 16*wave + r (+8 for lanes 16-31), N = lane&15
        const int col = nt * 16 + colB;
        if (col < NN) {
            const int rbase = 16 * wave + ((lane >> 4) << 3);
#pragma unroll
            for (int r = 0; r < 8; ++r) {
                const int row = rbase + r;
                if (row < NN)
                    out_b[row * (2 * NN) + NN + col] = acc[r];
            }
        }
    }
}

extern "C" void kernel_launch(void* const* d_in, const int* in_sizes, int n_in,
                              void* d_out, int out_size, void* d_ws, size_t ws_size,
                              hipStream_t stream) {
    const float* x     = (const float*)d_in[0];   // (4096, 87, 87)
    const float* evals = (const float*)d_in[1];   // (4096, 87)
    const float* evecs = (const float*)d_in[2];   // (4096, 87, 87)
    const float* dtime = (const float*)d_in[3];   // (87,)
    float* out = (float*)d_out;                   // (4096, 87, 174)

    diffusion_kernel<<<NB, THREADS, 0, stream>>>(x, evals, evecs, dtime, out);
}